// SYNNETWORK_83288005804288
// MI455X (gfx1250) — compile-verified
//
#include <hip/hip_runtime.h>

typedef __attribute__((ext_vector_type(2))) float v2f;
typedef __attribute__((ext_vector_type(8))) float v8f;
typedef int v4i __attribute__((vector_size(16)));

typedef __attribute__((address_space(1))) v4i* gptr4_t;   // global int4*
typedef __attribute__((address_space(3))) v4i* lptr4_t;   // LDS int4*

#define B_      64
#define D_      10000
#define T_      200
#define O_      3
#define NCHUNK  25
#define DCHUNK  400      // D_ / NCHUNK, multiple of 8
#define TTILES  13       // ceil(T_ / 16)

#if __has_builtin(__builtin_amdgcn_global_load_async_to_lds_b128)
#define HAVE_ASYNC_LDS 1
#endif

// ---------------------------------------------------------------------------
// Kernel 1: per-(t-tile, batch, d-chunk) partial GEMM via V_WMMA_F32_16X16X4_F32
//   C[t, o] += A[t, k] * B[k, o],  A[t,k] = x[b, d0+k, t],  B[k,o] = W[o, d0+k]
// One wave per block; EXEC all-ones through the WMMA loop. Out-of-range t / o
// lanes use CLAMPED addresses only (no mask multiplies): they contaminate only
// C rows/cols that the epilogue never stores.
// ---------------------------------------------------------------------------
__global__ __launch_bounds__(32)
void fc_wmma_partial(const float* __restrict__ x,
                     const float* __restrict__ W,
                     float* __restrict__ partial) {
    // 1280 floats: 1200 useful (3 x 400) + 80 floats of pad for uniform async count
    __shared__ float lw[1280];

    const int tid  = threadIdx.x;              // 0..31
    const int half = tid >> 4;                 // lane group 0/1
    const int m    = tid & 15;                 // row/col within group
    const int t0   = blockIdx.x * 16;
    const int b    = blockIdx.y;
    const int c    = blockIdx.z;
    const int d0   = c * DCHUNK;

    // ---- Stage W chunk into LDS via async global->LDS B128 DMA (ASYNCcnt path).
    // 300 useful 16B chunks (3 rows x 100), padded to 320 => 10 per lane.
    for (int i = tid; i < 320; i += 32) {
        const int ic  = (i < 300) ? i : 299;   // clamp source for pad transfers
        const int o   = ic / 100;              // W row
        const int rem = ic - o * 100;          // 16B chunk within row
        const float* g = W + (size_t)o * D_ + d0 + rem * 4;
        float*       l = lw + i * 4;           // pad lands in lw[1200..1279]
#ifdef HAVE_ASYNC_LDS
        __builtin_amdgcn_global_load_async_to_lds_b128(
            (gptr4_t)(void*)g, (lptr4_t)(void*)l, 0, 0);
#else
        *(float4*)l = *(const float4*)g;
#endif
    }
#ifdef HAVE_ASYNC_LDS
    asm volatile("s_wait_asynccnt 0x0" ::: "memory");
#endif
    __syncthreads();

    // ---- Per-lane clamped base pointers (computed once; loop uses imm offsets).
    const int t   = t0 + m;
    const int tcl = (t < T_) ? t : (T_ - 1);   // clamp; rows t>=200 are discarded
    const int oc  = (m < O_) ? m : 0;          // clamp; cols o>=3 are discarded

    // A: lanes 0-15 hold K=+0/+1, lanes 16-31 hold K=+2/+3; M = lane % 16.
    const float* xp  = x + ((size_t)b * D_ + (d0 + 2 * half)) * T_ + tcl;
    // B: N = lane % 16, same K split by lane group.
    const float* lwp = lw + oc * DCHUNK + 2 * half;

    v8f acc0 = {};
    v8f acc1 = {};
#pragma unroll 2
    for (int k = 0; k < DCHUNK; k += 8) {
        v2f a0 = { xp[(size_t)k * T_],        xp[(size_t)k * T_ + T_] };
        v2f b0 = { lwp[k],                    lwp[k + 1] };
        acc0 = __builtin_amdgcn_wmma_f32_16x16x4_f32(false, a0, false, b0,
                                                     (short)0, acc0, false, false);
        v2f a1 = { xp[(size_t)(k + 4) * T_],  xp[(size_t)(k + 4) * T_ + T_] };
        v2f b1 = { lwp[k + 4],                lwp[k + 5] };
        acc1 = __builtin_amdgcn_wmma_f32_16x16x4_f32(false, a1, false, b1,
                                                     (short)0, acc1, false, false);
    }
    v8f acc = acc0 + acc1;

    // C/D layout: VGPR r -> (lanes 0-15: M=r, N=lane), (lanes 16-31: M=r+8).
    if (m < O_) {
#pragma unroll
        for (int r = 0; r < 8; ++r) {
            int tm = t0 + r + 8 * half;
            if (tm < T_) {
                partial[(((size_t)c * B_ + b) * T_ + tm) * O_ + m] = acc[r];
            }
        }
    }
}

// ---------------------------------------------------------------------------
// Kernel 2: reduce partials over the 25 d-chunks, add bias -> fc[t][b][o].
// Fixed summation order => deterministic.
// ---------------------------------------------------------------------------
__global__ void fc_reduce(const float* __restrict__ partial,
                          const float* __restrict__ bias,
                          float* __restrict__ fc) {
    int g = blockIdx.x * blockDim.x + threadIdx.x;
    if (g >= T_ * B_ * O_) return;
    int t = g / (B_ * O_);
    int j = g - t * (B_ * O_);
    int b = j / O_;
    int o = j - b * O_;
    float s = bias[o];
    for (int c = 0; c < NCHUNK; ++c)
        s += partial[(((size_t)c * B_ + b) * T_ + t) * O_ + o];
    fc[g] = s;
}

// ---------------------------------------------------------------------------
// Kernel 3: LIF scan (snntorch subtract-reset, Heaviside forward).
// 192 threads = 64 batches x 3 neurons, sequential over T=200.
// ---------------------------------------------------------------------------
__global__ __launch_bounds__(192)
void lif_scan(const float* __restrict__ fc,
              const float* __restrict__ alpha_p,
              const float* __restrict__ beta_p,
              const float* __restrict__ th_p,
              float* __restrict__ out) {
    const int   j     = threadIdx.x;           // b*3 + o
    const float alpha = alpha_p[0];
    const float beta  = beta_p[0];
    const float th    = th_p[0];

    float syn = 0.0f, mem = 0.0f;
    float* spk_rec = out;
    float* mem_rec = out + T_ * B_ * O_;

    for (int t = 0; t < T_; ++t) {
        float inp   = fc[t * (B_ * O_) + j];
        float reset = (mem > th) ? th : 0.0f;  // reset from PREVIOUS mem
        syn = alpha * syn + inp;
        mem = beta * mem + syn - reset;
        spk_rec[t * (B_ * O_) + j] = ((mem - th) > 0.0f) ? 1.0f : 0.0f;
        mem_rec[t * (B_ * O_) + j] = mem;
    }
}

// ---------------------------------------------------------------------------
extern "C" void kernel_launch(void* const* d_in, const int* in_sizes, int n_in,
                              void* d_out, int out_size, void* d_ws, size_t ws_size,
                              hipStream_t stream) {
    const float* x  = (const float*)d_in[0];   // [64,100,100,200]
    const float* W  = (const float*)d_in[1];   // [3,10000]
    const float* bi = (const float*)d_in[2];   // [3]
    const float* al = (const float*)d_in[3];   // alpha
    const float* be = (const float*)d_in[4];   // beta
    const float* th = (const float*)d_in[5];   // threshold
    float*       out = (float*)d_out;          // spk_rec | mem_rec, 2*38400 f32

    float* partial = (float*)d_ws;                                  // 25*64*200*3 f32
    float* fc      = partial + (size_t)NCHUNK * B_ * T_ * O_;       // 38400 f32

    dim3 g1(TTILES, B_, NCHUNK);
    fc_wmma_partial<<<g1, 32, 0, stream>>>(x, W, partial);

    const int n = T_ * B_ * O_;
    fc_reduce<<<(n + 255) / 256, 256, 0, stream>>>(partial, bi, fc);

    lif_scan<<<1, 192, 0, stream>>>(fc, al, be, th, out);
}